// SparseLatentAddressRead_2714419331663
// MI455X (gfx1250) — compile-verified
//
#include <hip/hip_runtime.h>
#include <math.h>

typedef __attribute__((ext_vector_type(2))) float v2f;
typedef __attribute__((ext_vector_type(8))) float v8f;
typedef int v4i __attribute__((vector_size(4 * sizeof(int))));

#define B_ 32
#define S_ 8192
#define D_ 64
#define A_ 32
#define N_ 64
#define K_ 8
#define INV_TEMP 4.0f

#define WAVES 8
#define TOKENS_PER_WG 1024
#define ITERS (TOKENS_PER_WG / (WAVES * 16)) // 8

#define WMMA_F32(a, b, c) \
    __builtin_amdgcn_wmma_f32_16x16x4_f32(false, (a), false, (b), (short)0, (c), false, false)

#if __has_builtin(__builtin_amdgcn_global_load_async_to_lds_b128) && \
    __has_builtin(__builtin_amdgcn_s_wait_asynccnt)
#define HAS_ASYNC_LDS 1
#else
#define HAS_ASYNC_LDS 0
#endif

#if HAS_ASYNC_LDS
#define GPTR(p) ((__attribute__((address_space(1))) v4i*)(p))
#define LPTR(p) ((__attribute__((address_space(3))) v4i*)(p))
// copy one 4 KB X tile (1024 contiguous floats) with 8 async-to-LDS ops
// (32 lanes x 16 B = 512 B per instruction); tracked by ASYNCcnt
__device__ __forceinline__ void async_tile_copy(const float* src, float* dst, int lane) {
    #pragma unroll
    for (int j = 0; j < 8; ++j) {
        __builtin_amdgcn_global_load_async_to_lds_b128(
            GPTR(src + j * 128 + lane * 4),
            LPTR(dst + j * 128 + lane * 4),
            0, 0);
    }
}
#endif

__global__ __launch_bounds__(256)
void slar_kernel(const float* __restrict__ x,
                 const float* __restrict__ mem_addr,
                 const float* __restrict__ mem_val,
                 const float* __restrict__ Wq,
                 const float* __restrict__ Wo,
                 float* __restrict__ out)
{
    // ---- shared tables (per workgroup) ----
    __shared__ float sWq[A_ * D_];     // Wq[a][d]       : B-operand for q-proj
    __shared__ float sWo[D_ * D_];     // Wo[e][d]       : B-operand for out-proj
    __shared__ float sAddr[N_ * A_];   // normalized addresses [n][a]
    __shared__ float sMem[N_ * D_];    // memory_values[b][n][d]
    // ---- per-wave staging ----
    // double-buffered 16x64 tile: holds X during q-proj, then is reused for
    // raw scores (16xN) and the READ tile (16xD) of the SAME iteration while
    // the async engine fills the other buffer with the next X tile.
    __shared__ float sXbuf[WAVES][2][16 * D_];
    __shared__ float sQ[WAVES][16 * A_];   // raw queries
    __shared__ float sTw[WAVES][16][K_];   // softmax weights
    __shared__ int   sTi[WAVES][16][K_];   // top-k indices

    const int tid  = threadIdx.x;
    const int w    = tid >> 5;
    const int lane = tid & 31;
    const int hl   = lane & 15;  // row (A) or col (B/C) within 16
    const int kh   = lane >> 4;  // K-half selector for A/B operands
    const int b    = blockIdx.y;

    // ================= one-time cooperative loads =================
    for (int i = tid; i < A_ * D_; i += 256) sWq[i] = Wq[i];
    for (int i = tid; i < D_ * D_; i += 256) sWo[i] = Wo[i];
    {
        const float4* mv  = reinterpret_cast<const float4*>(mem_val + (size_t)b * N_ * D_);
        float4*       dst = reinterpret_cast<float4*>(sMem);
        for (int i = tid; i < (N_ * D_) / 4; i += 256) dst[i] = mv[i];
    }
    if (tid < N_) {  // l2-normalize one address row per thread
        float tmp[A_];
        float ss = 0.f;
        #pragma unroll
        for (int a = 0; a < A_; ++a) { tmp[a] = mem_addr[tid * A_ + a]; ss += tmp[a] * tmp[a]; }
        const float inv = 1.0f / fmaxf(sqrtf(ss), 1e-12f);
        #pragma unroll
        for (int a = 0; a < A_; ++a) sAddr[tid * A_ + a] = tmp[a] * inv;
    }
    __syncthreads();

    const int base = blockIdx.x * TOKENS_PER_WG;
    const float* xw = x + ((size_t)b * S_ + base + w * 16) * D_;  // this wave's tile 0

#if HAS_ASYNC_LDS
    // prime the pipeline: async-DMA tile 0 into buffer 0
    async_tile_copy(xw, sXbuf[w][0], lane);
#endif

    for (int it = 0; it < ITERS; ++it) {
        const int s0 = base + (it * WAVES + w) * 16;   // this wave's 16-token tile
        float* xb = sXbuf[w][it & 1];                  // X -> scores -> READ staging

#if HAS_ASYNC_LDS
        // issue next tile's DMA first, then wait for the current tile only
        if (it + 1 < ITERS) {
            async_tile_copy(xw + (size_t)(it + 1) * WAVES * 16 * D_,
                            sXbuf[w][(it + 1) & 1], lane);
            __builtin_amdgcn_s_wait_asynccnt(8);   // current tile's 8 ops done
        } else {
            __builtin_amdgcn_s_wait_asynccnt(0);
        }
#else
        const float* xt = xw + (size_t)it * WAVES * 16 * D_;
        if (it + 1 < ITERS) {   // global_prefetch_b8 for next tile
            const char* nx = (const char*)(xt + (size_t)WAVES * 16 * D_);
            __builtin_prefetch(nx + lane * 128, 0, 1);
        }
        {   // synchronous stage: global -> VGPR -> LDS
            const float4* src = reinterpret_cast<const float4*>(xt);
            float4*       dst = reinterpret_cast<float4*>(xb);
            #pragma unroll
            for (int j = 0; j < 8; ++j) dst[j * 32 + lane] = src[j * 32 + lane];
        }
        __syncthreads();
#endif

        // ================= q = X @ Wq^T : [16xD]@[DxA] =================
        #pragma unroll
        for (int nt = 0; nt < A_ / 16; ++nt) {
            v8f c = {0.f, 0.f, 0.f, 0.f, 0.f, 0.f, 0.f, 0.f};
            #pragma unroll
            for (int kk = 0; kk < D_ / 4; ++kk) {
                v2f a  = *(const v2f*)&xb[hl * D_ + kk * 4 + kh * 2];
                v2f bb = *(const v2f*)&sWq[(nt * 16 + hl) * D_ + kk * 4 + kh * 2];
                c = WMMA_F32(a, bb, c);
            }
            #pragma unroll
            for (int r = 0; r < 8; ++r)
                sQ[w][(r + 8 * kh) * A_ + nt * 16 + hl] = c[r];
        }
        __syncthreads();

        // ==== scores = Q @ addrN^T : [16xA]@[AxN] -> overwrite X buffer ====
        #pragma unroll
        for (int nt = 0; nt < N_ / 16; ++nt) {
            v8f c = {0.f, 0.f, 0.f, 0.f, 0.f, 0.f, 0.f, 0.f};
            #pragma unroll
            for (int kk = 0; kk < A_ / 4; ++kk) {
                v2f a  = *(const v2f*)&sQ[w][hl * A_ + kk * 4 + kh * 2];
                v2f bb = *(const v2f*)&sAddr[(nt * 16 + hl) * A_ + kk * 4 + kh * 2];
                c = WMMA_F32(a, bb, c);
            }
            #pragma unroll
            for (int r = 0; r < 8; ++r)
                xb[(r + 8 * kh) * N_ + nt * 16 + hl] = c[r];
        }
        __syncthreads();

        // ====== top-8 + softmax; q-norm folded into score scale ======
        if (lane < 16) {
            const int t = lane;
            float ss = 0.f;
            #pragma unroll
            for (int a = 0; a < A_; ++a) { const float v = sQ[w][t * A_ + a]; ss += v * v; }
            const float scale = INV_TEMP / fmaxf(sqrtf(ss), 1e-12f);

            float tv[K_]; int ti[K_];
            #pragma unroll
            for (int i = 0; i < K_; ++i) { tv[i] = -3.4e38f; ti[i] = 0; }
            for (int n = 0; n < N_; ++n) {
                const float v = xb[t * N_ + n] * scale;
                if (v > tv[K_ - 1]) {          // strict > keeps earlier-index ties first
                    int p = K_ - 1;
                    while (p > 0 && v > tv[p - 1]) { tv[p] = tv[p - 1]; ti[p] = ti[p - 1]; --p; }
                    tv[p] = v; ti[p] = n;
                }
            }
            const float m = tv[0];
            float e[K_]; float sum = 0.f;
            #pragma unroll
            for (int i = 0; i < K_; ++i) { e[i] = __expf(tv[i] - m); sum += e[i]; }
            const float rs = 1.0f / sum;
            #pragma unroll
            for (int i = 0; i < K_; ++i) { sTw[w][t][i] = e[i] * rs; sTi[w][t][i] = ti[i]; }
        }
        __syncthreads();

        // ====== read = sum_k weight*mem[idx] -> same buffer (scores dead) ======
        {
            const int t = hl, dh = kh;        // 32 lanes = 16 tokens x 2 half-rows
            float wgt[K_]; int off[K_];
            #pragma unroll
            for (int i = 0; i < K_; ++i) {
                wgt[i] = sTw[w][t][i];
                off[i] = sTi[w][t][i] * D_ + dh * 32;
            }
            for (int d = 0; d < 32; ++d) {
                float acc = 0.f;
                #pragma unroll
                for (int i = 0; i < K_; ++i) acc += wgt[i] * sMem[off[i] + d];
                xb[t * D_ + dh * 32 + d] = acc;
            }
        }
        __syncthreads();

        // ============ out = READ @ Wo^T : [16xD]@[DxD] -> global ============
        float* ot = out + ((size_t)b * S_ + s0) * D_;
        #pragma unroll
        for (int nt = 0; nt < D_ / 16; ++nt) {
            v8f c = {0.f, 0.f, 0.f, 0.f, 0.f, 0.f, 0.f, 0.f};
            #pragma unroll
            for (int kk = 0; kk < D_ / 4; ++kk) {
                v2f a  = *(const v2f*)&xb[hl * D_ + kk * 4 + kh * 2];
                v2f bb = *(const v2f*)&sWo[(nt * 16 + hl) * D_ + kk * 4 + kh * 2];
                c = WMMA_F32(a, bb, c);
            }
            #pragma unroll
            for (int r = 0; r < 8; ++r)
                ot[(size_t)(r + 8 * kh) * D_ + nt * 16 + hl] = c[r];
        }
        __syncthreads();
    }
}

extern "C" void kernel_launch(void* const* d_in, const int* in_sizes, int n_in,
                              void* d_out, int out_size, void* d_ws, size_t ws_size,
                              hipStream_t stream) {
    (void)in_sizes; (void)n_in; (void)out_size; (void)d_ws; (void)ws_size;
    const float* x  = (const float*)d_in[0];
    const float* ma = (const float*)d_in[1];
    const float* mv = (const float*)d_in[2];
    const float* Wq = (const float*)d_in[3];
    const float* Wo = (const float*)d_in[4];
    float* out = (float*)d_out;

    dim3 grid(S_ / TOKENS_PER_WG, B_);   // (8, 32) workgroups
    slar_kernel<<<grid, 256, 0, stream>>>(x, ma, mv, Wq, Wo, out);
}